// G2InvariantPooling_43224550867441
// MI455X (gfx1250) — compile-verified
//
#include <hip/hip_runtime.h>
#include <stdint.h>
#include <math.h>

// ---------------------------------------------------------------------------
// G2 invariant pooling: per point (7-dim x,y,z) -> 10 features.
// Memory bound (~1.5 flop/byte; ~520 MB total traffic -> ~22us floor at
// 23.3 TB/s). Strategy: stream tiles Global->LDS with CDNA5 async loads
// (global_load_async_to_lds_b64, non-temporal), double buffered via
// s_wait_asynccnt, compute per-point features from LDS, non-temporal
// coalesced float2 stores. All data is touched exactly once -> NT policy
// end-to-end so L2 isn't churned.
// ---------------------------------------------------------------------------

#define TPB            256
#define TILE_PTS       512                       // points per tile
#define FLOATS_PER_ARR (TILE_PTS * 7)            // 3584 floats
#define BYTES_PER_ARR  (FLOATS_PER_ARR * 4)      // 14336 B
#define FLOATS_PER_BUF (3 * FLOATS_PER_ARR)      // 10752 floats
#define BYTES_PER_BUF  (FLOATS_PER_BUF * 4)      // 43008 B
#define LDS_BYTES      (2 * BYTES_PER_BUF)       // 86016 B (dynamic shared)
// Async loads per wave per tile: 3 arrays * 7 chunks = 21 (per-wave ASYNCcnt)

typedef __attribute__((ext_vector_type(2))) float f32x2;

__device__ __forceinline__ void async_ld_b64(uint32_t lds_byte, uint32_t gl_byte,
                                             const void* sbase) {
  // GVS mode: saddr (SGPR64 base) + 32-bit per-lane byte offset.
  // VDST holds the per-lane LDS byte address. Tracked by ASYNCcnt.
  // TH_LOAD_NT: streamed exactly once, don't retain in caches.
  asm volatile("global_load_async_to_lds_b64 %0, %1, %2 th:TH_LOAD_NT"
               :
               : "v"(lds_byte), "v"(gl_byte), "s"(sbase)
               : "memory");
}

__device__ __forceinline__ void prefetch_tile(const float* __restrict__ x,
                                              const float* __restrict__ y,
                                              const float* __restrict__ z,
                                              long long tile,
                                              uint32_t lds_buf_byte,
                                              uint32_t t) {
  const float* xb = x + (size_t)tile * FLOATS_PER_ARR;
  const float* yb = y + (size_t)tile * FLOATS_PER_ARR;
  const float* zb = z + (size_t)tile * FLOATS_PER_ARR;
#pragma unroll
  for (int k = 0; k < 7; ++k) {
    // thread t moves float2 chunk (t + k*256): wave moves 256 contiguous bytes
    uint32_t o = t * 8u + (uint32_t)k * (TPB * 8u);
    async_ld_b64(lds_buf_byte + o,                      o, (const void*)xb);
    async_ld_b64(lds_buf_byte + BYTES_PER_ARR + o,      o, (const void*)yb);
    async_ld_b64(lds_buf_byte + 2u * BYTES_PER_ARR + o, o, (const void*)zb);
  }
}

// G2 associative 3-form phi (Bryant): e123+e145+e167+e246-e257-e347-e356 (1-based)
// cross_k = phi_ijk x_i y_j ; triple = <cross, z>
__device__ __forceinline__ void g2_features(const float X[7], const float Y[7],
                                            const float Z[7], float f[10]) {
  float xx = 0.f, xy = 0.f, yy = 0.f, xz = 0.f, yz = 0.f, zz = 0.f;
#pragma unroll
  for (int i = 0; i < 7; ++i) {
    xx = fmaf(X[i], X[i], xx);
    xy = fmaf(X[i], Y[i], xy);
    yy = fmaf(Y[i], Y[i], yy);
    xz = fmaf(X[i], Z[i], xz);
    yz = fmaf(Y[i], Z[i], yz);
    zz = fmaf(Z[i], Z[i], zz);
  }
  const float c0 = (X[1]*Y[2]-X[2]*Y[1]) + (X[3]*Y[4]-X[4]*Y[3]) + (X[5]*Y[6]-X[6]*Y[5]);
  const float c1 = (X[2]*Y[0]-X[0]*Y[2]) + (X[3]*Y[5]-X[5]*Y[3]) - (X[4]*Y[6]-X[6]*Y[4]);
  const float c2 = (X[0]*Y[1]-X[1]*Y[0]) - (X[3]*Y[6]-X[6]*Y[3]) - (X[4]*Y[5]-X[5]*Y[4]);
  const float c3 = (X[4]*Y[0]-X[0]*Y[4]) + (X[5]*Y[1]-X[1]*Y[5]) - (X[6]*Y[2]-X[2]*Y[6]);
  const float c4 = (X[0]*Y[3]-X[3]*Y[0]) - (X[6]*Y[1]-X[1]*Y[6]) - (X[5]*Y[2]-X[2]*Y[5]);
  const float c5 = (X[6]*Y[0]-X[0]*Y[6]) + (X[1]*Y[3]-X[3]*Y[1]) - (X[2]*Y[4]-X[4]*Y[2]);
  const float c6 = (X[0]*Y[5]-X[5]*Y[0]) - (X[1]*Y[4]-X[4]*Y[1]) - (X[2]*Y[3]-X[3]*Y[2]);

  float cc = c0*c0;
  cc = fmaf(c1, c1, cc); cc = fmaf(c2, c2, cc); cc = fmaf(c3, c3, cc);
  cc = fmaf(c4, c4, cc); cc = fmaf(c5, c5, cc); cc = fmaf(c6, c6, cc);

  float tri = c0*Z[0];
  tri = fmaf(c1, Z[1], tri); tri = fmaf(c2, Z[2], tri); tri = fmaf(c3, Z[3], tri);
  tri = fmaf(c4, Z[4], tri); tri = fmaf(c5, Z[5], tri); tri = fmaf(c6, Z[6], tri);

  f[0] = sqrtf(xx);   // ||x||
  f[1] = xy;          // <x,y>
  f[2] = sqrtf(yy);   // ||y||
  f[3] = sqrtf(cc);   // ||x x y||
  f[4] = xz;          // <x,z>
  f[5] = yz;          // <y,z>
  f[6] = sqrtf(zz);   // ||z||
  f[7] = tri;         // phi(x,y,z)
  f[8] = 0.f;         // phi(x,x,y) == 0
  f[9] = 0.f;         // phi(y,y,z) == 0
}

__device__ __forceinline__ void store_features_nt(float* __restrict__ out,
                                                  size_t pg, const float f[10]) {
  f32x2* o = reinterpret_cast<f32x2*>(out + pg * 10);  // 40B*p is 8-aligned
  f32x2 v;
  v.x = f[0]; v.y = f[1]; __builtin_nontemporal_store(v, o + 0);
  v.x = f[2]; v.y = f[3]; __builtin_nontemporal_store(v, o + 1);
  v.x = f[4]; v.y = f[5]; __builtin_nontemporal_store(v, o + 2);
  v.x = f[6]; v.y = f[7]; __builtin_nontemporal_store(v, o + 3);
  v.x = f[8]; v.y = f[9]; __builtin_nontemporal_store(v, o + 4);
}

__global__ __launch_bounds__(TPB)
void g2_pipeline_kernel(const float* __restrict__ x, const float* __restrict__ y,
                        const float* __restrict__ z, float* __restrict__ out,
                        long long ntiles) {
  extern __shared__ float lds[];
  const uint32_t t    = threadIdx.x;
  // Low 32 bits of a flat pointer into LDS == LDS byte offset (ISA 10.2)
  const uint32_t lds0 = (uint32_t)(uintptr_t)&lds[0];
  const long long stride = (long long)gridDim.x;

  long long tile = (long long)blockIdx.x;   // grid <= ntiles, so tile valid
  int buf = 0;
  prefetch_tile(x, y, z, tile, lds0, t);    // 21 async loads outstanding

  while (true) {
    const long long nxt = tile + stride;
    const bool have_next = (nxt < ntiles);
    if (have_next) {
      prefetch_tile(x, y, z, nxt, lds0 + (uint32_t)((buf ^ 1) * BYTES_PER_BUF), t);
      // counter == 42 in-order: wait until only next tile's 21 remain
      asm volatile("s_wait_asynccnt 21" ::: "memory");
    } else {
      asm volatile("s_wait_asynccnt 0" ::: "memory");
    }
    __syncthreads();   // current buffer fully resident for all waves

    const float* lx = lds + buf * FLOATS_PER_BUF;
    const float* ly = lx + FLOATS_PER_ARR;
    const float* lz = ly + FLOATS_PER_ARR;

#pragma unroll
    for (int pp = 0; pp < TILE_PTS / TPB; ++pp) {
      const int pl = (int)t + pp * TPB;     // local point index
      float X[7], Y[7], Z[7];
#pragma unroll
      for (int f = 0; f < 7; ++f) {         // addr = pl*7+f: gcd(7,64)=1 -> no bank conflicts
        X[f] = lx[pl * 7 + f];
        Y[f] = ly[pl * 7 + f];
        Z[f] = lz[pl * 7 + f];
      }
      float f[10];
      g2_features(X, Y, Z, f);

      const size_t pg = (size_t)tile * TILE_PTS + (size_t)pl;
      store_features_nt(out, pg, f);
    }
    __syncthreads();   // everyone done reading buf before it is overwritten
    if (!have_next) break;
    tile = nxt;
    buf ^= 1;
  }
}

// Generic tail (remainder points), direct global loads. Not used when
// npts % TILE_PTS == 0 (the reference shape is exactly divisible).
__global__ __launch_bounds__(TPB)
void g2_tail_kernel(const float* __restrict__ x, const float* __restrict__ y,
                    const float* __restrict__ z, float* __restrict__ out,
                    long long start_pt, long long npts) {
  const long long p = start_pt + (long long)blockIdx.x * TPB + threadIdx.x;
  if (p >= npts) return;
  float X[7], Y[7], Z[7];
#pragma unroll
  for (int f = 0; f < 7; ++f) {
    X[f] = x[p * 7 + f];
    Y[f] = y[p * 7 + f];
    Z[f] = z[p * 7 + f];
  }
  float ftr[10];
  g2_features(X, Y, Z, ftr);
  store_features_nt(out, (size_t)p, ftr);
}

extern "C" void kernel_launch(void* const* d_in, const int* in_sizes, int n_in,
                              void* d_out, int out_size, void* d_ws, size_t ws_size,
                              hipStream_t stream) {
  const float* x = (const float*)d_in[0];
  const float* y = (const float*)d_in[1];
  const float* z = (const float*)d_in[2];
  float* out = (float*)d_out;

  const long long nelem  = (long long)in_sizes[0];   // npts * 7
  const long long npts   = nelem / 7;                // 4,194,304 for ref shape
  const long long ntiles = npts / TILE_PTS;          // 8192
  const long long tail0  = ntiles * TILE_PTS;

  if (ntiles > 0) {
    long long grid = ntiles < 2048 ? ntiles : 2048;  // persistent, double-buffered
    g2_pipeline_kernel<<<dim3((unsigned)grid), dim3(TPB), LDS_BYTES, stream>>>(
        x, y, z, out, ntiles);
  }
  if (tail0 < npts) {
    const long long ntail = npts - tail0;
    const int tg = (int)((ntail + TPB - 1) / TPB);
    g2_tail_kernel<<<dim3(tg), dim3(TPB), 0, stream>>>(x, y, z, out, tail0, npts);
  }
}